// DetectionHead_43971875176515
// MI455X (gfx1250) — compile-verified
//
#include <hip/hip_runtime.h>

typedef float v4f __attribute__((ext_vector_type(4)));

#define TX   64      // tile width  (outputs)
#define TY   16      // tile height (outputs)
#define PS   68      // padded LDS row stride (16B-aligned rows, incl. 2 slack cols)
#define PYR  18      // padded rows
#define PN   (PS * PYR)   // 1224 floats per LDS tile
#define IMW  1024
#define IMH  1024
#define NBATCH 16
#define EPSV 0.01f

// Self-probe for the CDNA5 async global->LDS path (ASYNCcnt). Confirmed present
// on this toolchain in round 1 (prototype: (as1 int*, as3 int*, imm offset, cpol)).
#if defined(__has_builtin)
#if __has_builtin(__builtin_amdgcn_global_load_async_to_lds_b32)
#define HAVE_ASYNC 1
#endif
#endif
#ifndef HAVE_ASYNC
#define HAVE_ASYNC 0
#endif

#if HAVE_ASYNC
// AS1 (global) non-const int*; C-style cast strips const and retypes.
#define GLBP(p) ((__attribute__((address_space(1))) int*)(p))
// AS3 (LDS) int*
#define LDSP(p) ((__attribute__((address_space(3))) int*)(p))
#endif

__global__ __launch_bounds__(256) void nms_fused_kernel(const float* __restrict__ seg,
                                                        float* __restrict__ out) {
  __shared__ __align__(16) float dt[PN];          // diff tile (padded, zero halo OOB)
#if HAVE_ASYNC
  __shared__ __align__(16) float t0[PN];          // raw channel-0 tile
  __shared__ __align__(16) float t1[PN];          // raw channel-1 tile
#endif

  const int tid = threadIdx.x;
  const int bx = blockIdx.x, by = blockIdx.y, b = blockIdx.z;
  const int x0 = bx * TX, y0 = by * TY;
  const size_t HW     = (size_t)IMH * IMW;
  const size_t plane0 = (size_t)(2 * b) * HW;
  const size_t plane1 = plane0 + HW;
  const size_t NX     = (size_t)NBATCH * HW;      // size of xnms output

  // ---- passthrough-copy source address (one float4 per channel per thread) ----
  const int crow = tid >> 4;                      // 0..15
  const int ccol = (tid & 15) << 2;               // 0,4,...,60
  const size_t ci = (size_t)(y0 + crow) * IMW + (size_t)(x0 + ccol);
  __builtin_prefetch(seg + plane0 + ci, 0, 3);    // global_prefetch_b8
  __builtin_prefetch(seg + plane1 + ci, 0, 3);

  // ---- fill padded tile ----
#if HAVE_ASYNC
  const bool border = (bx == 0) | (by == 0) | (bx == (IMW / TX - 1)) | (by == (IMH / TY - 1));
  if (border) {
    // pre-zero so EXEC-masked (out-of-image) lanes leave zeros behind
    for (int e = tid; e < PN; e += 256) { t0[e] = 0.0f; t1[e] = 0.0f; }
    __syncthreads();                              // waits DScnt before async writes land
  }
  for (int e = tid; e < PN; e += 256) {
    const int py = e / PS;
    const int px = e - py * PS;
    const int gy = y0 + py - 1;
    const int gx = x0 + px - 1;
    if ((unsigned)gy < (unsigned)IMH && (unsigned)gx < (unsigned)IMW) {
      const size_t gi = (size_t)gy * IMW + (size_t)gx;
      __builtin_amdgcn_global_load_async_to_lds_b32(GLBP(seg + plane0 + gi),
                                                    LDSP(&t0[e]), 0, 0);
      __builtin_amdgcn_global_load_async_to_lds_b32(GLBP(seg + plane1 + gi),
                                                    LDSP(&t1[e]), 0, 0);
    }
  }
#else
  for (int e = tid; e < PN; e += 256) {
    const int py = e / PS;
    const int px = e - py * PS;
    const int gy = y0 + py - 1;
    const int gx = x0 + px - 1;
    float d = 0.0f;
    if ((unsigned)gy < (unsigned)IMH && (unsigned)gx < (unsigned)IMW) {
      const size_t gi = (size_t)gy * IMW + (size_t)gx;
      d = seg[plane1 + gi] - seg[plane0 + gi] - EPSV;
    }
    dt[e] = d;
  }
#endif

  // ---- passthrough copy (overlaps with in-flight async fill) ----
  {
    const v4f c0v = *(const v4f*)(seg + plane0 + ci);
    const v4f c1v = *(const v4f*)(seg + plane1 + ci);
    __builtin_nontemporal_store(c0v, (v4f*)(out + NX + plane0 + ci));
    __builtin_nontemporal_store(c1v, (v4f*)(out + NX + plane1 + ci));
  }

#if HAVE_ASYNC
#if __has_builtin(__builtin_amdgcn_s_wait_asynccnt)
  __builtin_amdgcn_s_wait_asynccnt(0);
#else
  asm volatile("s_wait_asynccnt 0" ::: "memory");
#endif
  __syncthreads();
  for (int e = tid; e < PN; e += 256)
    dt[e] = t1[e] - t0[e] - EPSV;
#endif
  __syncthreads();

  // ---- 8-neighbor hole max of relu(diff), compare, NT store 4 outputs ----
  const int r  = tid >> 4;                        // output row in tile, 0..15
  const int c0 = (tid & 15) << 2;                 // first output col in tile, padded col base
  const float* pr0 = &dt[(r + 0) * PS + c0];
  const float* pr1 = &dt[(r + 1) * PS + c0];
  const float* pr2 = &dt[(r + 2) * PS + c0];
  const v4f a0 = *(const v4f*)(pr0);
  const v4f a1 = *(const v4f*)(pr0 + 4);          // overread into slack cols (unused lanes)
  const v4f b0 = *(const v4f*)(pr1);
  const v4f b1 = *(const v4f*)(pr1 + 4);
  const v4f g0 = *(const v4f*)(pr2);
  const v4f g1 = *(const v4f*)(pr2 + 4);

  const float pa[6] = { fmaxf(a0.x, 0.0f), fmaxf(a0.y, 0.0f), fmaxf(a0.z, 0.0f),
                        fmaxf(a0.w, 0.0f), fmaxf(a1.x, 0.0f), fmaxf(a1.y, 0.0f) };
  const float pb[6] = { b0.x, b0.y, b0.z, b0.w, b1.x, b1.y };
  const float pg[6] = { fmaxf(g0.x, 0.0f), fmaxf(g0.y, 0.0f), fmaxf(g0.z, 0.0f),
                        fmaxf(g0.w, 0.0f), fmaxf(g1.x, 0.0f), fmaxf(g1.y, 0.0f) };

  v4f o;
#pragma unroll
  for (int j = 0; j < 4; ++j) {
    const float am = fmaxf(fmaxf(pa[j], pa[j + 1]), pa[j + 2]);
    const float gm = fmaxf(fmaxf(pg[j], pg[j + 1]), pg[j + 2]);
    const float bm = fmaxf(fmaxf(pb[j], 0.0f), fmaxf(pb[j + 2], 0.0f));
    const float xm = fmaxf(fmaxf(am, gm), bm);
    const float x  = pb[j + 1];
    o[j] = (x > xm) ? x : 0.0f;
  }
  __builtin_nontemporal_store(
      o, (v4f*)(out + ((size_t)b * IMH + (size_t)(y0 + r)) * IMW + (size_t)(x0 + c0)));
}

extern "C" void kernel_launch(void* const* d_in, const int* in_sizes, int n_in,
                              void* d_out, int out_size, void* d_ws, size_t ws_size,
                              hipStream_t stream) {
  (void)in_sizes; (void)n_in; (void)out_size; (void)d_ws; (void)ws_size;
  const float* seg = (const float*)d_in[0];      // [16, 2, 1024, 1024] fp32
  float* out = (float*)d_out;                    // [16M xnms | 32M segmentation copy]
  dim3 grid(IMW / TX, IMH / TY, NBATCH);         // (16, 64, 16)
  nms_fused_kernel<<<grid, 256, 0, stream>>>(seg, out);
}